// Loss_80917183856990
// MI455X (gfx1250) — compile-verified
//
#include <hip/hip_runtime.h>
#include <hip/hip_bf16.h>

// YOLO loss for MI455X (gfx1250, wave32).
// Kernel 1: one block per image (8 wave32 waves). Box fields staged to LDS via
//           CDNA5 async global->LDS loads (ASYNCcnt). Sparse per-target IoU +
//           gathered class loss. Deterministic fixed-order reductions.
// Kernel 2: batch reduction of 1024 per-image partials via chained
//           V_WMMA_F32_16X16X4_F32 (ones^T * A * ones), then /B.

#define SGRID   14
#define NCELL   196          // 14*14
#define NCH     95           // 5 + 90
#define NUM_CLS 90
#define NWAVES  8

typedef float v2f __attribute__((ext_vector_type(2)));
typedef float v8f __attribute__((ext_vector_type(8)));

#if defined(__has_builtin)
#if __has_builtin(__builtin_amdgcn_global_load_async_to_lds_b32)
#define HAS_ASYNC_LDS 1
#endif
#if __has_builtin(__builtin_amdgcn_wmma_f32_16x16x4_f32)
#define HAS_WMMA_F32 1
#endif
#endif

typedef __attribute__((address_space(1))) int g_int;
typedef __attribute__((address_space(3))) int l_int;

__device__ __forceinline__ void g2l_b32(const float* g, float* l) {
#ifdef HAS_ASYNC_LDS
    __builtin_amdgcn_global_load_async_to_lds_b32(
        (g_int*)g, (l_int*)l, 0, 0);
#else
    *l = *g;
#endif
}

__device__ __forceinline__ void wait_async_lds() {
#ifdef HAS_ASYNC_LDS
#if __has_builtin(__builtin_amdgcn_s_wait_asynccnt)
    __builtin_amdgcn_s_wait_asynccnt(0);
#else
    asm volatile("s_wait_asynccnt 0" ::: "memory");
#endif
#endif
}

__global__ __launch_bounds__(256)
void yolo_loss_per_image(const float* __restrict__ P,
                         const float* __restrict__ G,
                         float* __restrict__ ws) {
    const int b    = blockIdx.x;
    const int tid  = threadIdx.x;
    const int lane = tid & 31;
    const int wave = tid >> 5;

    __shared__ float sp[5][NCELL];   // pred  x,y,w,h,conf
    __shared__ float sg[5][NCELL];   // targ  x,y,w,h,conf
    __shared__ float stx[NCELL], sty[NCELL];
    __shared__ float svf[NCELL];
    __shared__ float shit[NCELL];
    __shared__ float sloss[256];
    __shared__ int   sHasObj;

    const size_t imgBase = (size_t)b * NCELL * NCH;

    if (tid == 0) sHasObj = 0;

    // ---- Phase 1: stage box fields (async global -> LDS, per-lane b32) ----
    if (tid < NCELL) {
        const float* pc = P + imgBase + (size_t)tid * NCH;
        const float* gc = G + imgBase + (size_t)tid * NCH;
#pragma unroll
        for (int c = 0; c < 5; ++c) g2l_b32(pc + c, &sp[c][tid]);
#pragma unroll
        for (int c = 0; c < 5; ++c) g2l_b32(gc + c, &sg[c][tid]);
    }
    wait_async_lds();
    __syncthreads();

    // ---- Phase 2: validity, global-coord target xy ----
    if (tid < NCELL) {
        const float row = (float)(tid / SGRID);
        const float col = (float)(tid % SGRID);
        const float vf  = (sg[4][tid] > 0.0f) ? 1.0f : 0.0f;
        svf[tid]  = vf;
        stx[tid]  = col / (float)SGRID + sg[0][tid] / (float)SGRID;
        sty[tid]  = row / (float)SGRID + sg[1][tid] / (float)SGRID;
        shit[tid] = 0.0f;
        if (vf > 0.0f) sHasObj = 1;          // benign same-value race
    }
    sloss[tid] = 0.0f;
    __syncthreads();

    // ---- Phase 3: one wave per valid target cell ----
    for (int n = wave; n < NCELL; n += NWAVES) {
        if (svf[n] == 0.0f) continue;        // uniform per wave
        const float txv = stx[n], tyv = sty[n];
        const float twv = sg[2][n], thv = sg[3][n];
        const float tArea = twv * thv;

        // IoU max+argmax over 196 preds, lanes stride 32; first-max tie-break.
        float bestV = -1.0f; int bestI = 0;
        for (int p = lane; p < NCELL; p += 32) {
            const float dx = fabsf(txv - sp[0][p]);
            const float dy = fabsf(tyv - sp[1][p]);
            const float pw = sp[2][p], ph = sp[3][p];
            const float sw = (twv + pw) * 0.5f;
            const float sh = (thv + ph) * 0.5f;
            const float inter = fmaxf((sw - dx) * (sh - dy), 0.0f);
            const float uni   = tArea + pw * ph - inter;
            const float iou   = inter / uni;
            if (iou > bestV) { bestV = iou; bestI = p; }
        }
        for (int off = 16; off > 0; off >>= 1) {
            const float ov = __shfl_xor(bestV, off, 32);
            const int   oi = __shfl_xor(bestI, off, 32);
            if (ov > bestV || (ov == bestV && oi < bestI)) { bestV = ov; bestI = oi; }
        }

        // Class SSE for this cell only (sparse gather from HBM).
        const float* pcls = P + imgBase + (size_t)n * NCH + 5;
        const float* gcls = G + imgBase + (size_t)n * NCH + 5;
        float cs = 0.0f;
        for (int c = lane; c < NUM_CLS; c += 32) {
            const float d = pcls[c] - gcls[c];
            cs += d * d;
        }
        for (int off = 16; off > 0; off >>= 1) cs += __shfl_xor(cs, off, 32);

        if (lane == 0) {
            const float bx = sp[0][bestI], by = sp[1][bestI], bc = sp[4][bestI];
            const float ex = bx - txv, ey = by - tyv;
            const float xy = ex * ex + ey * ey;
            const float ec = bc * bestV - 1.0f;
            // total = LC*(xy+xy) + con_obj + LN*nocon + cls ; LC=5 -> 10*xy
            sloss[n] = 10.0f * xy + ec * ec + cs;
            shit[bestI] = 1.0f;              // benign same-value race
        }
    }
    __syncthreads();

    // ---- Phase 4: no-object confidence loss (only if image has objects) ----
    if (tid < NCELL && sHasObj) {
        const float c = sp[4][tid];
        sloss[tid] += 0.5f * (1.0f - shit[tid]) * c * c;
    }
    __syncthreads();

    // ---- Fixed-order tree reduction over 256 slots (deterministic) ----
    for (int s = 128; s > 0; s >>= 1) {
        if (tid < s) sloss[tid] += sloss[tid + s];
        __syncthreads();
    }
    if (tid == 0) ws[b] = sloss[0];
}

// Final reduction: sum M per-image partials with chained f32 WMMA
// (D = A x Ones + C accumulates row-sums; deterministic HW order), /M.
__global__ __launch_bounds__(32)
void yolo_loss_reduce(const float* __restrict__ ws, float* __restrict__ out, int M) {
    const int lane = threadIdx.x;
#ifdef HAS_WMMA_F32
    v8f acc = {0.f, 0.f, 0.f, 0.f, 0.f, 0.f, 0.f, 0.f};
    const v2f ones = {1.0f, 1.0f};
    // A-matrix 16x4 f32 layout: lanes 0-15 hold K=0,1 ; lanes 16-31 hold K=2,3.
    const int base_off = 4 * (lane & 15) + 2 * (lane >> 4);   // even -> 8B aligned
    if ((M & 63) == 0) {
        // Fast path (B is a multiple of 64): unpredicated b64 loads, no
        // divergence anywhere near the WMMA chain (EXEC stays all-ones).
        const v2f* wv = (const v2f*)ws;
        const int vbase = base_off >> 1;
        const int chunks = M >> 6;
        for (int i = 0; i < chunks; ++i) {
            const v2f a = wv[i * 32 + vbase];
            acc = __builtin_amdgcn_wmma_f32_16x16x4_f32(
                false, a, false, ones, (short)0, acc, false, false);
        }
    } else {
        const int chunks = (M + 63) >> 6;
        for (int i = 0; i < chunks; ++i) {
            const int j = i * 64 + base_off;
            v2f a;
            a.x = (j     < M) ? ws[j]     : 0.0f;
            a.y = (j + 1 < M) ? ws[j + 1] : 0.0f;
            acc = __builtin_amdgcn_wmma_f32_16x16x4_f32(
                false, a, false, ones, (short)0, acc, false, false);
        }
    }
    // Column N=0 of D: lane 0 holds M=0..7, lane 16 holds M=8..15.
    float s = acc[0] + acc[1] + acc[2] + acc[3] + acc[4] + acc[5] + acc[6] + acc[7];
    const float tot = __shfl(s, 0, 32) + __shfl(s, 16, 32);
    if (lane == 0) out[0] = tot / (float)M;
#else
    if (lane == 0) {
        float t = 0.0f;
        for (int i = 0; i < M; ++i) t += ws[i];
        out[0] = t / (float)M;
    }
#endif
}

extern "C" void kernel_launch(void* const* d_in, const int* in_sizes, int n_in,
                              void* d_out, int out_size, void* d_ws, size_t ws_size,
                              hipStream_t stream) {
    const float* P = (const float*)d_in[0];   // p_yolo [B,14,14,95] f32
    const float* G = (const float*)d_in[1];   // g_yolo [B,14,14,95] f32
    float* out = (float*)d_out;               // scalar f32
    float* ws  = (float*)d_ws;                // B per-image partials

    const int Bc = in_sizes[0] / (NCELL * NCH);
    yolo_loss_per_image<<<Bc, 256, 0, stream>>>(P, G, ws);
    yolo_loss_reduce<<<1, 32, 0, stream>>>(ws, out, Bc);
}